// AutoCorrelation_10909216932584
// MI455X (gfx1250) — compile-verified
//
#include <hip/hip_runtime.h>
#include <math.h>

typedef __attribute__((ext_vector_type(2))) float v2f;
typedef __attribute__((ext_vector_type(4))) float v4f;
typedef __attribute__((ext_vector_type(8))) float v8f;

#define B_  8
#define T_  2048
#define E_  128
#define H_  4
#define NQ  512          // E*H, inner width of q/k/v
#define BT  16384        // B*T
#define TOPK 22          // int(3*log(2048))

// ---------------------------------------------------------------------------
// Kernel 1: fused QKV projection.  P = X @ W + bias  for one of q/k/v
//   X  : (BT, 128)  row-major
//   W  : (128, 512) row-major (== wq/wk/wv (E,E,H) flat)
//   out: (BT, 512)  row-major (== q/k/v (B,T,E,H) flat)
// grid (BT/128, 512/128, 3), block 256 (8 waves); wave = 16 rows x 128 cols.
// fp32 WMMA 16x16x4, K loop of 32 steps.
// ---------------------------------------------------------------------------
__global__ __launch_bounds__(256) void qkv_gemm_kernel(
    const float* __restrict__ X,
    const float* __restrict__ Wq, const float* __restrict__ Bq,
    const float* __restrict__ Wk, const float* __restrict__ Bk,
    const float* __restrict__ Wv, const float* __restrict__ Bv,
    float* __restrict__ Q, float* __restrict__ K, float* __restrict__ V)
{
    const int wave = threadIdx.x >> 5;
    const int lane = threadIdx.x & 31;
    const int half = lane >> 4;     // 0: K={0,1}, 1: K={2,3}
    const int l16  = lane & 15;

    const float* W; const float* Bias; float* Out;
    if (blockIdx.z == 0)      { W = Wq; Bias = Bq; Out = Q; }
    else if (blockIdx.z == 1) { W = Wk; Bias = Bk; Out = K; }
    else                      { W = Wv; Bias = Bv; Out = V; }

    const int m0 = blockIdx.x * 128 + wave * 16;   // 16 output rows per wave
    const int n0 = blockIdx.y * 128;               // 128 output cols per block

    v8f acc[8];
    #pragma unroll
    for (int i = 0; i < 8; ++i) { v8f z = {}; acc[i] = z; }

    const int arow = m0 + l16;                     // A-matrix row for this lane
    const float* __restrict__ xrow = X + arow * E_;

    for (int kk = 0; kk < E_; kk += 4) {
        const int ca = kk + 2 * half;              // this lane's K pair
        v2f a;
        a.x = xrow[ca];
        a.y = xrow[ca + 1];
        #pragma unroll
        for (int ng = 0; ng < 8; ++ng) {
            const int n = n0 + ng * 16 + l16;      // B-matrix column
            v2f bf;
            bf.x = W[ca * NQ + n];
            bf.y = W[(ca + 1) * NQ + n];
            acc[ng] = __builtin_amdgcn_wmma_f32_16x16x4_f32(
                false, a, false, bf, (short)0, acc[ng], false, false);
        }
    }

    #pragma unroll
    for (int ng = 0; ng < 8; ++ng) {
        const int n = n0 + ng * 16 + l16;
        const float bias = Bias[n];
        #pragma unroll
        for (int r = 0; r < 8; ++r) {
            const int m = m0 + r + 8 * half;       // C/D layout: M = r + 8*half
            Out[m * NQ + n] = acc[ng][r] + bias;
        }
    }
}

// ---------------------------------------------------------------------------
// Kernel 2: circular autocorrelation mean over heads.
//   ac_mean[b,u,tau] = (1/H) * sum_t  q[b,(t+tau)%T,u,:] . k[b,t,u,:]
// grid (T/256, 128, B), block 256; one tau per thread.
// Dynamic LDS (starts at byte 0): 512-entry sliding q window (float4) at 0,
// 256-entry k chunk at byte 8192.  Staged with GLOBAL_LOAD_ASYNC_TO_LDS_B128
// (ASYNCcnt-tracked, GV mode), fenced with s_wait_asynccnt + barrier.
// Inner loop uses 4-wide ext-vector math -> packed fp32 FMA (VOP3P).
// ---------------------------------------------------------------------------
__global__ __launch_bounds__(256) void corr_kernel(
    const float* __restrict__ Q, const float* __restrict__ K,
    float* __restrict__ AC)                         // (B,128,T)
{
    extern __shared__ v4f sm[];                     // [0,512): qs, [512,768): ks

    const int tid  = threadIdx.x;
    const int b    = blockIdx.z;
    const int u    = blockIdx.y;
    const int tau0 = blockIdx.x * 256;
    const int tau  = tau0 + tid;

    // float4 granules: element (b,t,u) lives at float index ((b*T+t)*E+u)*4
    const float* __restrict__ Qb = Q + (size_t)b * T_ * NQ + u * 4;
    const float* __restrict__ Kb = K + (size_t)b * T_ * NQ + u * 4;

    v4f acc = {};

    for (int tc = 0; tc < T_; tc += 256) {
        __syncthreads();                            // previous chunk consumed

        // --- async stage: k chunk (one b128 per lane) ---
        {
            const unsigned lds_off = 8192u + 16u * (unsigned)tid;   // ks[tid]
            const unsigned long long ga =
                (unsigned long long)(Kb + (size_t)(tc + tid) * NQ);
            asm volatile("global_load_async_to_lds_b128 %0, %1, off"
                         :: "v"(lds_off), "v"(ga) : "memory");
        }
        // --- async stage: q window, 512 entries (two b128 per lane) ---
        {
            const int j0 = tid;
            const int j1 = tid + 256;
            const unsigned lo0 = 16u * (unsigned)j0;
            const unsigned lo1 = 16u * (unsigned)j1;
            const unsigned long long ga0 = (unsigned long long)
                (Qb + (size_t)((tc + tau0 + j0) & (T_ - 1)) * NQ);
            const unsigned long long ga1 = (unsigned long long)
                (Qb + (size_t)((tc + tau0 + j1) & (T_ - 1)) * NQ);
            asm volatile("global_load_async_to_lds_b128 %0, %1, off"
                         :: "v"(lo0), "v"(ga0) : "memory");
            asm volatile("global_load_async_to_lds_b128 %0, %1, off"
                         :: "v"(lo1), "v"(ga1) : "memory");
        }

        // prefetch next k chunk toward L2 while we compute
        if (tc + 256 < T_)
            __builtin_prefetch(Kb + (size_t)(tc + 256 + tid) * NQ, 0, 0);

        asm volatile("s_wait_asynccnt 0x0" ::: "memory");
        __syncthreads();                            // all waves' tiles visible

        #pragma unroll 8
        for (int tp = 0; tp < 256; ++tp) {
            const v4f kv = sm[512 + tp];            // broadcast across lanes
            const v4f qv = sm[tp + tid];            // conflict-free slide
            acc += qv * kv;                         // packed fp32 FMA
        }
    }
    AC[(b * E_ + u) * T_ + tau] =
        0.25f * (acc.x + acc.y + acc.z + acc.w);
}

// ---------------------------------------------------------------------------
// Kernel 3: top-22 (iterative argmax, lower-index tie-break) + softmax tail.
// grid (B*128), block 256.  Emits delay = idx[21], wgt = softmax(vals)[21].
// ---------------------------------------------------------------------------
__global__ __launch_bounds__(256) void topk_kernel(
    const float* __restrict__ AC, int* __restrict__ Delay,
    float* __restrict__ Wgt)
{
    __shared__ float vals[T_];
    __shared__ float rv[256];
    __shared__ int   ri[256];
    __shared__ float topv[TOPK];
    __shared__ int   topi[TOPK];

    const int bu  = blockIdx.x;
    const int tid = threadIdx.x;

    for (int i = tid; i < T_; i += 256) vals[i] = AC[bu * T_ + i];
    __syncthreads();

    for (int r = 0; r < TOPK; ++r) {
        float best = -__builtin_inff();
        int   bi   = T_;
        for (int i = tid; i < T_; i += 256) {
            const float v = vals[i];
            if (v > best) { best = v; bi = i; }
        }
        rv[tid] = best; ri[tid] = bi;
        __syncthreads();
        for (int s = 128; s > 0; s >>= 1) {
            if (tid < s) {
                const float v2 = rv[tid + s];
                const int   i2 = ri[tid + s];
                if (v2 > rv[tid] || (v2 == rv[tid] && i2 < ri[tid])) {
                    rv[tid] = v2; ri[tid] = i2;
                }
            }
            __syncthreads();
        }
        if (tid == 0) {
            topv[r] = rv[0];
            topi[r] = ri[0];
            vals[ri[0]] = -__builtin_inff();
        }
        __syncthreads();
    }

    if (tid == 0) {
        const float vmax = topv[0];
        float s = 0.f;
        #pragma unroll
        for (int i = 0; i < TOPK; ++i) s += __expf(topv[i] - vmax);
        Delay[bu] = topi[TOPK - 1];
        Wgt[bu]   = __expf(topv[TOPK - 1] - vmax) / s;
    }
}

// ---------------------------------------------------------------------------
// Kernel 4: output projection with gathered/rolled/scaled A operand.
//   out[b,t,f] = sum_{u,h} v[b,(t+delay[b,u])%T,u,h]*wgt[b,u]*wo[(u*4+h)*128+f]
//              + bo[f]
// grid (T/128, 1, B), block 256; fp32 WMMA, M=t, N=f(128), K=512.
// ---------------------------------------------------------------------------
__global__ __launch_bounds__(256) void out_gemm_kernel(
    const float* __restrict__ V, const int* __restrict__ Delay,
    const float* __restrict__ Wgt, const float* __restrict__ Wo,
    const float* __restrict__ Bo, float* __restrict__ Out)
{
    const int wave = threadIdx.x >> 5;
    const int lane = threadIdx.x & 31;
    const int half = lane >> 4;
    const int l16  = lane & 15;

    const int b  = blockIdx.z;
    const int t0 = blockIdx.x * 128 + wave * 16;
    const int trow = t0 + l16;                       // A-matrix row (time)

    v8f acc[8];
    #pragma unroll
    for (int i = 0; i < 8; ++i) { v8f z = {}; acc[i] = z; }

    for (int kk = 0; kk < NQ; kk += 4) {
        const int ca = kk + 2 * half;                // K pair; h in {0,2}
        const int u  = ca >> 2;
        const int h  = ca & 3;
        const int d  = Delay[b * E_ + u];
        const float wg = Wgt[b * E_ + u];
        const int ts = (trow + d) & (T_ - 1);
        const float* __restrict__ vp =
            V + ((b * T_ + ts) * E_ + u) * H_ + h;   // h, h+1 contiguous
        v2f a;
        a.x = vp[0] * wg;
        a.y = vp[1] * wg;
        #pragma unroll
        for (int ng = 0; ng < 8; ++ng) {
            const int n = ng * 16 + l16;
            v2f bf;
            bf.x = Wo[ca * E_ + n];
            bf.y = Wo[(ca + 1) * E_ + n];
            acc[ng] = __builtin_amdgcn_wmma_f32_16x16x4_f32(
                false, a, false, bf, (short)0, acc[ng], false, false);
        }
    }

    #pragma unroll
    for (int ng = 0; ng < 8; ++ng) {
        const int n = ng * 16 + l16;
        const float bias = Bo[n];
        #pragma unroll
        for (int r = 0; r < 8; ++r) {
            const int t = t0 + r + 8 * half;
            Out[(b * T_ + t) * E_ + n] = acc[ng][r] + bias;
        }
    }
}

// ---------------------------------------------------------------------------
extern "C" void kernel_launch(void* const* d_in, const int* in_sizes, int n_in,
                              void* d_out, int out_size, void* d_ws, size_t ws_size,
                              hipStream_t stream) {
    const float* hs = (const float*)d_in[0];
    const float* wq = (const float*)d_in[1];
    const float* bq = (const float*)d_in[2];
    const float* wk = (const float*)d_in[3];
    const float* bk = (const float*)d_in[4];
    const float* wv = (const float*)d_in[5];
    const float* bv = (const float*)d_in[6];
    const float* wo = (const float*)d_in[7];
    const float* bo = (const float*)d_in[8];

    const size_t qkv_elems = (size_t)BT * NQ;        // 8,388,608 floats each
    float* q  = (float*)d_ws;
    float* k  = q  + qkv_elems;
    float* v  = k  + qkv_elems;
    float* ac = v  + qkv_elems;                      // B*E*T = 2,097,152 floats
    int*   dl = (int*)(ac + (size_t)B_ * E_ * T_);
    float* wg = (float*)(dl + B_ * E_);

    // 1) QKV projections (fp32 WMMA)
    dim3 g1(BT / 128, NQ / 128, 3);
    qkv_gemm_kernel<<<g1, 256, 0, stream>>>(hs, wq, bq, wk, bk, wv, bv, q, k, v);

    // 2) circular autocorrelation mean (async-LDS staged, packed fp32 FMA)
    dim3 g2(T_ / 256, E_, B_);
    const size_t corr_lds = (512 + 256) * sizeof(float4);   // 12 KB dynamic
    corr_kernel<<<g2, 256, corr_lds, stream>>>(q, k, ac);

    // 3) top-22 + softmax tail -> delay / weight
    topk_kernel<<<B_ * E_, 256, 0, stream>>>(ac, dl, wg);

    // 4) rolled+weighted output projection (fp32 WMMA)
    dim3 g4(T_ / 128, 1, B_);
    out_gemm_kernel<<<g4, 256, 0, stream>>>(v, dl, wg, wo, bo, (float*)d_out);
}